// CBAM_59992103190552
// MI455X (gfx1250) — compile-verified
//
#include <hip/hip_runtime.h>
#include <math.h>

// CBAM: channel attention (avg/max pool -> shared MLP -> sigmoid sum) then
// spatial attention (channel avg/max -> 7x7 conv -> sigmoid).
// x: [32, 56, 56, 256] fp32, NHWC.

#define CHN 256
#define HIDN 16
#define BAT 32
#define HGT 56
#define WID 56
#define HW  (HGT * WID)        // 3136
#define NSL 16                 // pixel slices per batch in partial reduce
#define SLICE (HW / NSL)       // 196
#define C4  (CHN / 4)          // 64 float4 per pixel

typedef float v2f __attribute__((ext_vector_type(2)));
typedef float v8f __attribute__((ext_vector_type(8)));

__device__ __forceinline__ float sig_(float v) {
    return __builtin_amdgcn_rcpf(1.0f + __expf(-v));   // v_rcp_f32, no div chain
}

// ---------------------------------------------------------------------------
// Kernel 1: partial channel pooling. Grid = 32 batches * 16 slices.
// 4 pixel-subgroups x 64 channel-quads; dense B128 loads, LDS combine.
// ---------------------------------------------------------------------------
__global__ __launch_bounds__(256) void k1_partial(const float* __restrict__ x,
                                                  float* __restrict__ psum,
                                                  float* __restrict__ pmax) {
    __shared__ float ssum[4 * CHN];   // [pixel-subgroup][channel]
    __shared__ float smax[4 * CHN];

    const int b = blockIdx.x >> 4;
    const int s = blockIdx.x & 15;
    const int t = threadIdx.x;
    const int pg = t >> 6;            // 0..3 pixel subgroup
    const int q  = t & 63;            // channel quad 0..63 (channels 4q..4q+3)

    const float4* xp = (const float4*)(x + ((size_t)b * HW + (size_t)s * SLICE) * CHN);

    float4 sum = {0.f, 0.f, 0.f, 0.f};
    float4 mx  = {-3.4e38f, -3.4e38f, -3.4e38f, -3.4e38f};
    for (int p = pg; p < SLICE; p += 4) {
        __builtin_prefetch(&xp[(size_t)(p + 16) * C4 + q], 0, 0);  // speculative
        float4 v = xp[(size_t)p * C4 + q];
        sum.x += v.x; sum.y += v.y; sum.z += v.z; sum.w += v.w;
        mx.x = fmaxf(mx.x, v.x); mx.y = fmaxf(mx.y, v.y);
        mx.z = fmaxf(mx.z, v.z); mx.w = fmaxf(mx.w, v.w);
    }
    ((float4*)ssum)[pg * C4 + q] = sum;
    ((float4*)smax)[pg * C4 + q] = mx;
    __syncthreads();

    const int c = t;                  // combine the 4 pixel subgroups
    float r = ssum[c] + ssum[CHN + c] + ssum[2 * CHN + c] + ssum[3 * CHN + c];
    float m = fmaxf(fmaxf(smax[c], smax[CHN + c]),
                    fmaxf(smax[2 * CHN + c], smax[3 * CHN + c]));
    psum[(size_t)blockIdx.x * CHN + c] = r;
    pmax[(size_t)blockIdx.x * CHN + c] = m;
}

// ---------------------------------------------------------------------------
// Kernel 2: finalize pools + shared MLP via V_WMMA_F32_16X16X4_F32.
// Single block, 256 threads = 8 waves (wave32).
//   GEMM1: pool[32x256] @ w1[256x16]  -> relu -> hid[32x16]   (waves 0-1)
//   GEMM2: hid [32x16 ] @ w2[16x256]  -> +b2 -> sigmoid       (8 waves x 4 tiles)
// gate = sigmoid(mlp(avg)) + sigmoid(mlp(max)), accumulated in v8f registers.
// ---------------------------------------------------------------------------
__global__ __launch_bounds__(256) void k2_gate(const float* __restrict__ psum,
                                               const float* __restrict__ pmax,
                                               const float* __restrict__ w1,
                                               const float* __restrict__ b1,
                                               const float* __restrict__ w2,
                                               const float* __restrict__ b2,
                                               float* __restrict__ gate) {
    __shared__ float pool[BAT][CHN];   // 32 KB
    __shared__ float hid[BAT][HIDN];   // 2 KB

    const int t = threadIdx.x;
    const int wave = t >> 5;
    const int lane = t & 31;
    const int l16 = lane & 15;         // N (or M) index within half-wave
    const int hi = lane >> 4;          // 0: lanes 0-15, 1: lanes 16-31

    v8f zero = {};
    v8f acc[4];
#pragma unroll
    for (int i = 0; i < 4; ++i) acc[i] = zero;

    for (int pass = 0; pass < 2; ++pass) {
        // --- Stage A: finalize avg (pass 0) or max (pass 1) pooling into LDS
        const float* part = pass ? pmax : psum;
        for (int b = 0; b < BAT; ++b) {
            float r = pass ? -3.4e38f : 0.0f;
            for (int s = 0; s < NSL; ++s) {
                float v = part[(size_t)(b * NSL + s) * CHN + t];
                r = pass ? fmaxf(r, v) : (r + v);
            }
            pool[b][t] = pass ? r : r * (1.0f / (float)HW);
        }
        __syncthreads();

        // --- Stage B: GEMM1 (M=32,K=256,N=16), waves 0-1 own one 16-row tile
        if (wave < 2) {
            const int m0 = wave * 16;
            v8f c = zero;
            for (int k = 0; k < CHN; k += 4) {
                // A 16x4 fp32: lanes 0-15 -> K = k,k+1 ; lanes 16-31 -> K = k+2,k+3
                v2f a, bm;
                a.x = pool[m0 + l16][k + 2 * hi + 0];
                a.y = pool[m0 + l16][k + 2 * hi + 1];
                // B 4x16 fp32: row K striped across lanes, mirrored split
                bm.x = w1[(k + 2 * hi + 0) * HIDN + l16];
                bm.y = w1[(k + 2 * hi + 1) * HIDN + l16];
                c = __builtin_amdgcn_wmma_f32_16x16x4_f32(false, a, false, bm,
                                                          (short)0, c, false, false);
            }
            const float bias = b1[l16];
#pragma unroll
            for (int r = 0; r < 8; ++r) {  // D: VGPR r -> M = m0 + r (+8 for lanes>=16)
                hid[m0 + r + 8 * hi][l16] = fmaxf(c[r] + bias, 0.0f);
            }
        }
        __syncthreads();

        // --- Stage C: GEMM2 (M=32,K=16,N=256): 2x16 = 32 tiles, 4 per wave
#pragma unroll
        for (int tt = 0; tt < 4; ++tt) {
            const int tile = wave * 4 + tt;
            const int m0 = (tile >> 4) * 16;
            const int n0 = (tile & 15) * 16;
            v8f c = zero;
#pragma unroll
            for (int k = 0; k < HIDN; k += 4) {
                v2f a, bm;
                a.x = hid[m0 + l16][k + 2 * hi + 0];
                a.y = hid[m0 + l16][k + 2 * hi + 1];
                bm.x = w2[(k + 2 * hi + 0) * CHN + n0 + l16];
                bm.y = w2[(k + 2 * hi + 1) * CHN + n0 + l16];
                c = __builtin_amdgcn_wmma_f32_16x16x4_f32(false, a, false, bm,
                                                          (short)0, c, false, false);
            }
            const float bias = b2[n0 + l16];
#pragma unroll
            for (int r = 0; r < 8; ++r) {
                acc[tt][r] += sig_(c[r] + bias);
            }
        }
        __syncthreads();   // hid/pool reused next pass
    }

    // --- Write gate[32][256]
#pragma unroll
    for (int tt = 0; tt < 4; ++tt) {
        const int tile = wave * 4 + tt;
        const int m0 = (tile >> 4) * 16;
        const int n0 = (tile & 15) * 16;
#pragma unroll
        for (int r = 0; r < 8; ++r) {
            gate[(size_t)(m0 + r + 8 * hi) * CHN + n0 + l16] = acc[tt][r];
        }
    }
}

// ---------------------------------------------------------------------------
// Kernel 3: spatial stats of gated x. One wave per pixel (8 pixels / block).
// Lane owns channel quads `lane` and `lane+32` -> two dense B128 loads.
// s[b,h,w,0] = mean_c(x*g), s[b,h,w,1] = max_c(x*g).  x read should hit L2.
// ---------------------------------------------------------------------------
__global__ __launch_bounds__(256) void k3_spatial(const float* __restrict__ x,
                                                  const float* __restrict__ gate,
                                                  float* __restrict__ sbuf) {
    const int wave = threadIdx.x >> 5;
    const int lane = threadIdx.x & 31;
    const int gp = blockIdx.x * 8 + wave;      // global pixel: 0 .. 32*3136-1
    const int b = gp / HW;
    const float4* xp = (const float4*)(x + (size_t)gp * CHN);
    const float4* g4 = (const float4*)(gate + (size_t)b * CHN);

    float sum = 0.0f, mx = -3.4e38f;
#pragma unroll
    for (int i = 0; i < 2; ++i) {
        const int qi = lane + 32 * i;          // dense across the wave
        const float4 xv = xp[qi];
        const float4 gv = g4[qi];
        const float p0 = xv.x * gv.x, p1 = xv.y * gv.y;
        const float p2 = xv.z * gv.z, p3 = xv.w * gv.w;
        sum += p0 + p1 + p2 + p3;
        mx = fmaxf(mx, fmaxf(fmaxf(p0, p1), fmaxf(p2, p3)));
    }
#pragma unroll
    for (int off = 16; off > 0; off >>= 1) {   // wave32 butterfly reduce
        sum += __shfl_xor(sum, off, 32);
        mx = fmaxf(mx, __shfl_xor(mx, off, 32));
    }
    if (lane == 0) {
        sbuf[(size_t)gp * 2 + 0] = sum * (1.0f / (float)CHN);
        sbuf[(size_t)gp * 2 + 1] = mx;
    }
}

// ---------------------------------------------------------------------------
// Kernel 4: 7x7 conv ("SAME") on s -> sigmoid -> final gated output.
// One block per (batch, row). Threads 0-55 each do one pixel's conv; then all
// threads stream the row with B128: fixed channel-quad, 4 pixels / iteration.
// ---------------------------------------------------------------------------
__global__ __launch_bounds__(256) void k4_out(const float* __restrict__ x,
                                              const float* __restrict__ gate,
                                              const float* __restrict__ sbuf,
                                              const float* __restrict__ cw,
                                              const float* __restrict__ cb,
                                              float* __restrict__ out) {
    __shared__ float sg[WID];
    const int b = blockIdx.x / HGT;
    const int h = blockIdx.x % HGT;
    const int t = threadIdx.x;

    if (t < WID) {
        float acc = cb[0];
        for (int dy = -3; dy <= 3; ++dy) {
            const int hh = h + dy;
            if (hh < 0 || hh >= HGT) continue;
            for (int dx = -3; dx <= 3; ++dx) {
                const int ww = t + dx;
                if (ww < 0 || ww >= WID) continue;
                const float* sp = sbuf + ((size_t)(b * HGT + hh) * WID + ww) * 2;
                const float* wp = cw + ((dy + 3) * 7 + (dx + 3)) * 2;  // HWIO
                acc += sp[0] * wp[0] + sp[1] * wp[1];
            }
        }
        sg[t] = sig_(acc);
    }
    __syncthreads();

    const int q    = t & 63;          // fixed channel quad (channels 4q..4q+3)
    const int wofs = t >> 6;          // pixel offset 0..3
    const float4 g = ((const float4*)gate)[(size_t)b * C4 + q];

    const size_t row = ((size_t)b * HGT + h) * WID;
    const float4* xr4 = (const float4*)(x + row * CHN);
    float4*       o4  = (float4*)(out + row * CHN);

#pragma unroll 2
    for (int i = 0; i < WID / 4; ++i) {        // 14 iterations, dense B128
        const int w = i * 4 + wofs;
        const float4 xv = xr4[(size_t)w * C4 + q];
        const float sgw = sg[w];
        float4 ov;
        ov.x = xv.x * g.x * sgw;
        ov.y = xv.y * g.y * sgw;
        ov.z = xv.z * g.z * sgw;
        ov.w = xv.w * g.w * sgw;
        o4[(size_t)w * C4 + q] = ov;
    }
}

// ---------------------------------------------------------------------------
extern "C" void kernel_launch(void* const* d_in, const int* in_sizes, int n_in,
                              void* d_out, int out_size, void* d_ws, size_t ws_size,
                              hipStream_t stream) {
    const float* x  = (const float*)d_in[0];
    const float* w1 = (const float*)d_in[1];
    const float* b1 = (const float*)d_in[2];
    const float* w2 = (const float*)d_in[3];
    const float* b2 = (const float*)d_in[4];
    const float* cw = (const float*)d_in[5];
    const float* cb = (const float*)d_in[6];
    float* out = (float*)d_out;

    // Workspace layout (floats): psum | pmax | gate | sbuf  (~1.85 MB total)
    float* ws   = (float*)d_ws;
    float* psum = ws;                                   // 32*16*256 = 131072
    float* pmax = ws + (size_t)BAT * NSL * CHN;         // 131072
    float* gate = pmax + (size_t)BAT * NSL * CHN;       // 32*256 = 8192
    float* sbuf = gate + (size_t)BAT * CHN;             // 32*56*56*2 = 200704

    k1_partial<<<BAT * NSL, 256, 0, stream>>>(x, psum, pmax);
    k2_gate<<<1, 256, 0, stream>>>(psum, pmax, w1, b1, w2, b2, gate);
    k3_spatial<<<BAT * HW / 8, 256, 0, stream>>>(x, gate, sbuf);
    k4_out<<<BAT * HGT, 256, 0, stream>>>(x, gate, sbuf, cw, cb, out);
}